// Voxelization_27023934226796
// MI455X (gfx1250) — compile-verified
//
#include <hip/hip_runtime.h>
#include <hip/hip_bf16.h>
#include <climits>

// ---------------------------------------------------------------------------
// Deterministic hard voxelization for gfx1250 (MI455X).
// Memory/atomic-bound problem: no matrix ops, so no WMMA. CDNA5-specific
// paths used: wave32 ballot compaction, async global->LDS staging
// (ASYNCcnt), global prefetch, LDS scans.
// ---------------------------------------------------------------------------

namespace vx {
constexpr int   C    = 5;
constexpr int   GX   = 1440, GY = 1440, GZ = 40;
constexpr float VXs  = 0.075f, VYs = 0.075f, VZs = 0.2f;
constexpr float X0   = -54.0f, Y0  = -54.0f, Z0  = -5.0f;
constexpr int   MAXV = 120000, MAXP = 10;
constexpr unsigned HBITS = 22u, HSIZE = 1u << HBITS, HMASK = HSIZE - 1u;
constexpr unsigned KEMPTY = 0xFFFFFFFFu;
// output layout (concatenated flat, float32):
constexpr int VOX_N   = MAXV * MAXP * C;      // 6,000,000 voxels elements
constexpr int COORS_O = VOX_N;                // 360,000 coors (z,y,x)
constexpr int NP_O    = COORS_O + MAXV * 3;   // 120,000 num_points
constexpr int VN_O    = NP_O + MAXV;          // 1 voxel_num
constexpr int OUT_N   = VN_O + 1;             // 6,480,001
}

__device__ __forceinline__ bool p2lin(float x, float y, float z,
                                      int& cx, int& cy, int& cz, int& lin) {
  cx = (int)floorf((x - vx::X0) / vx::VXs);
  cy = (int)floorf((y - vx::Y0) / vx::VYs);
  cz = (int)floorf((z - vx::Z0) / vx::VZs);
  if (cx < 0 || cy < 0 || cz < 0 || cx >= vx::GX || cy >= vx::GY || cz >= vx::GZ)
    return false;
  lin = cz * (vx::GX * vx::GY) + cy * vx::GX + cx;
  return true;
}

// ---- K0: initialize maps, ranks, and zero the output buffer ----------------
__global__ void k_init(unsigned* __restrict__ key, int* __restrict__ firstval,
                       int* __restrict__ slotmap, int* __restrict__ ranks,
                       float* __restrict__ out) {
  for (int i = blockIdx.x * blockDim.x + threadIdx.x; i < vx::OUT_N;
       i += gridDim.x * blockDim.x) {
    if (i < (int)vx::HSIZE) {
      key[i]      = vx::KEMPTY;
      firstval[i] = INT_MAX;
      slotmap[i]  = INT_MAX;
    }
    if (i < vx::MAXV * vx::MAXP) ranks[i] = INT_MAX;
    out[i] = 0.0f;
  }
}

// ---- K1: hash-insert each point's voxel, atomicMin first-occurrence --------
// Streams xyz through LDS using the CDNA5 async global->LDS path.
__global__ void k_assign(const float* __restrict__ pts, int n,
                         unsigned* __restrict__ key, int* __restrict__ firstval,
                         int* __restrict__ pentry) {
  // Sole LDS object in this kernel -> allocated at LDS offset 0, which the
  // async-to-LDS instructions address directly (a VGPR holds the LDS byte offset).
  __shared__ float tile[256 * 3];
  const int i = blockIdx.x * 256 + (int)threadIdx.x;
  if (i < n) {
    const unsigned goff = (unsigned)i * 20u;            // byte offset of row i
    const unsigned loff = (unsigned)threadIdx.x * 12u;  // LDS byte offset
#pragma unroll
    for (int j = 0; j < 3; ++j) {
      asm volatile("global_load_async_to_lds_b32 %0, %1, %2 offset:0"
                   :: "v"(loff + 4u * (unsigned)j),
                      "v"(goff + 4u * (unsigned)j),
                      "s"(pts)
                   : "memory");
    }
    if (i + 256 < n) __builtin_prefetch(pts + (size_t)(i + 256) * 5, 0, 0);
  }
  asm volatile("s_wait_asynccnt 0x0" ::: "memory");
  if (i >= n) return;
  const float x = tile[threadIdx.x * 3 + 0];
  const float y = tile[threadIdx.x * 3 + 1];
  const float z = tile[threadIdx.x * 3 + 2];
  int cx, cy, cz, lin;
  if (!p2lin(x, y, z, cx, cy, cz, lin)) {
    pentry[i] = -1;
    return;
  }
  unsigned h = ((unsigned)lin * 2654435761u) & vx::HMASK;
  while (true) {
    unsigned prev = atomicCAS(&key[h], vx::KEMPTY, (unsigned)lin);
    if (prev == vx::KEMPTY || prev == (unsigned)lin) break;
    h = (h + 1u) & vx::HMASK;
  }
  atomicMin(&firstval[h], i);  // order-independent -> deterministic
  pentry[i] = (int)h;
}

// ---- K2: per-block leader counts (wave32 ballot + popcount) ----------------
__global__ void k_count(const int* __restrict__ firstval,
                        const int* __restrict__ pentry, int n,
                        int* __restrict__ bcount) {
  const int i = blockIdx.x * 256 + (int)threadIdx.x;
  bool leader = false;
  if (i < n) {
    int e  = pentry[i];
    leader = (e >= 0) && (firstval[e] == i);
  }
  unsigned m = __builtin_amdgcn_ballot_w32(leader);
  __shared__ int wsum[8];
  const int lane = threadIdx.x & 31, wid = threadIdx.x >> 5;
  if (lane == 0) wsum[wid] = __popc(m);
  __syncthreads();
  if (threadIdx.x == 0) {
    int s = 0;
#pragma unroll
    for (int w = 0; w < 8; ++w) s += wsum[w];
    bcount[blockIdx.x] = s;
  }
}

// ---- K3: single-block exclusive scan of block counts; writes voxel_num -----
__global__ void k_scan(const int* __restrict__ bcount, int* __restrict__ boff,
                       int nb, float* __restrict__ out) {
  __shared__ int sh[256];
  int carry = 0;
  for (int base = 0; base < nb; base += 256) {
    const int idx = base + (int)threadIdx.x;
    const int v   = (idx < nb) ? bcount[idx] : 0;
    sh[threadIdx.x] = v;
    __syncthreads();
    for (int o = 1; o < 256; o <<= 1) {
      int t = (threadIdx.x >= (unsigned)o) ? sh[threadIdx.x - o] : 0;
      __syncthreads();
      sh[threadIdx.x] += t;
      __syncthreads();
    }
    if (idx < nb) boff[idx] = carry + sh[threadIdx.x] - v;  // exclusive
    carry += sh[255];
    __syncthreads();
  }
  if (threadIdx.x == 0)
    out[vx::VN_O] = (float)(carry > vx::MAXV ? vx::MAXV : carry);
}

// ---- K4: assign creation-order slots to leaders; write coors + rank 0 ------
__global__ void k_slots(const float* __restrict__ pts, int n,
                        const int* __restrict__ firstval,
                        const int* __restrict__ pentry,
                        const int* __restrict__ boff,
                        int* __restrict__ slotmap, int* __restrict__ ranks,
                        float* __restrict__ out) {
  const int i = blockIdx.x * 256 + (int)threadIdx.x;
  bool leader = false;
  int e = -1;
  if (i < n) {
    e      = pentry[i];
    leader = (e >= 0) && (firstval[e] == i);
  }
  unsigned m = __builtin_amdgcn_ballot_w32(leader);
  __shared__ int wsum[8];
  const int lane = threadIdx.x & 31, wid = threadIdx.x >> 5;
  if (lane == 0) wsum[wid] = __popc(m);
  __syncthreads();
  if (!leader) return;
  int woff = 0;
  for (int w = 0; w < wid; ++w) woff += wsum[w];
  const int slot = boff[blockIdx.x] + woff + __popc(m & ((1u << lane) - 1u));
  slotmap[e] = slot;
  if (slot < vx::MAXV) {
    const float x = pts[(size_t)i * 5 + 0];
    const float y = pts[(size_t)i * 5 + 1];
    const float z = pts[(size_t)i * 5 + 2];
    int cx, cy, cz, lin;
    p2lin(x, y, z, cx, cy, cz, lin);  // leader point is always in range
    out[vx::COORS_O + slot * 3 + 0] = (float)cz;
    out[vx::COORS_O + slot * 3 + 1] = (float)cy;
    out[vx::COORS_O + slot * 3 + 2] = (float)cx;
    ranks[slot * vx::MAXP] = i;  // smallest index == first occurrence
  }
}

// ---- K5: rank pass r: rank[r] = min{ i : i > rank[r-1] } (deterministic) ---
__global__ void k_rank(const int* __restrict__ pentry,
                       const int* __restrict__ slotmap, int* __restrict__ ranks,
                       int n, int r) {
  const int i = blockIdx.x * 256 + (int)threadIdx.x;
  if (i >= n) return;
  const int e = pentry[i];
  if (e < 0) return;
  const int s = slotmap[e];
  if ((unsigned)s >= (unsigned)vx::MAXV) return;
  const int prev = ranks[s * vx::MAXP + r - 1];
  if (i > prev) atomicMin(&ranks[s * vx::MAXP + r], i);
}

// ---- K6: emit voxels (gather 5 floats per kept point) + num_points ---------
__global__ void k_emit(const float* __restrict__ pts, int n,
                       const int* __restrict__ ranks, float* __restrict__ out) {
  const int t = blockIdx.x * 256 + (int)threadIdx.x;
  if (t >= vx::MAXV * vx::MAXP) return;
  const int s = t / vx::MAXP;
  const int r = t - s * vx::MAXP;
  const int p = ranks[t];
  if (p < n) {
    const float* src = pts + (size_t)p * 5;
#pragma unroll
    for (int k = 0; k < 5; ++k) out[(size_t)t * 5 + k] = src[k];
  }
  if (r == 0) {
    int cnt = 0;
#pragma unroll
    for (int q = 0; q < vx::MAXP; ++q)
      cnt += (ranks[s * vx::MAXP + q] < n) ? 1 : 0;
    out[vx::NP_O + s] = (float)cnt;  // == min(group count, MAX_POINTS)
  }
}

// ---------------------------------------------------------------------------
extern "C" void kernel_launch(void* const* d_in, const int* in_sizes, int n_in,
                              void* d_out, int out_size, void* d_ws,
                              size_t ws_size, hipStream_t stream) {
  (void)n_in; (void)out_size; (void)ws_size;
  const float* pts = (const float*)d_in[0];
  const int n      = in_sizes[0] / vx::C;  // 2,000,000 points, 5 features
  float* out       = (float*)d_out;

  // Carve workspace (~61 MB), 256B-aligned regions.
  char*  ws  = (char*)d_ws;
  size_t off = 0;
  auto take = [&](size_t bytes) -> void* {
    void* p = ws + off;
    off += (bytes + 255) & ~(size_t)255;
    return p;
  };
  unsigned* key      = (unsigned*)take((size_t)vx::HSIZE * 4);
  int*      firstval = (int*)take((size_t)vx::HSIZE * 4);
  int*      slotmap  = (int*)take((size_t)vx::HSIZE * 4);
  int*      pentry   = (int*)take((size_t)(n > 0 ? n : 1) * 4);
  int*      ranks    = (int*)take((size_t)vx::MAXV * vx::MAXP * 4);
  const int nb       = (n + 255) / 256;
  int*      bcount   = (int*)take((size_t)(nb > 0 ? nb : 1) * 4);
  int*      boff     = (int*)take((size_t)(nb > 0 ? nb : 1) * 4);

  hipLaunchKernelGGL(k_init, dim3(4096), dim3(256), 0, stream,
                     key, firstval, slotmap, ranks, out);
  if (n > 0) {
    hipLaunchKernelGGL(k_assign, dim3(nb), dim3(256), 0, stream,
                       pts, n, key, firstval, pentry);
    hipLaunchKernelGGL(k_count, dim3(nb), dim3(256), 0, stream,
                       firstval, pentry, n, bcount);
    hipLaunchKernelGGL(k_scan, dim3(1), dim3(256), 0, stream,
                       bcount, boff, nb, out);
    hipLaunchKernelGGL(k_slots, dim3(nb), dim3(256), 0, stream,
                       pts, n, firstval, pentry, boff, slotmap, ranks, out);
    for (int r = 1; r < vx::MAXP; ++r)
      hipLaunchKernelGGL(k_rank, dim3(nb), dim3(256), 0, stream,
                         pentry, slotmap, ranks, n, r);
  }
  const int nt = vx::MAXV * vx::MAXP;
  hipLaunchKernelGGL(k_emit, dim3((nt + 255) / 256), dim3(256), 0, stream,
                     pts, n, ranks, out);
}